// MAETrimLoss_66640712564888
// MI455X (gfx1250) — compile-verified
//
#include <hip/hip_runtime.h>
#include <hip/hip_bf16.h>
#include <stdint.h>

// ---------------------------------------------------------------------------
// Trimmed-MAE loss via radix-select (no sort).
//   For each batch row: T = u_m-th smallest |pred-target|;
//   loss_b = (sum_{v<T} v + (u_m - count_{v<T}) * T) / (2m);  out = mean_b.
// Radix select on float bit patterns (|x|>=0 => uint order == float order):
//   3 histogram levels: bits[31:21] (2048), bits[20:10] (2048), bits[9:0] (1024)
// Inputs stay L2-resident (78.6MB << 192MB), so 4 streaming passes are cheap.
// Scans are one wave32 per batch (segment sums + shfl_up prefix + owner walk).
// ---------------------------------------------------------------------------

typedef __attribute__((ext_vector_type(2))) float v2f;
typedef __attribute__((ext_vector_type(8))) float v8f;

constexpr int      B_SZ  = 32;
constexpr int      M_SZ  = 480 * 640;              // 307200 elements per row
constexpr unsigned K_SEL = 245760u;                // int(0.8 * M_SZ)
constexpr int      BLOCKS_PER_BATCH = 75;
constexpr int      CHUNK = M_SZ / BLOCKS_PER_BATCH; // 4096
constexpr int      THREADS = 256;

// workspace layout, in 32-bit words
constexpr size_t HIST0_OFF  = 0;                                   // 32*2048
constexpr size_t HIST1_OFF  = HIST0_OFF + (size_t)B_SZ * 2048;     // 32*2048
constexpr size_t HIST2_OFF  = HIST1_OFF + (size_t)B_SZ * 2048;     // 32*1024
constexpr size_t PREFIX_OFF = HIST2_OFF + (size_t)B_SZ * 1024;     // 32
constexpr size_t CLESS_OFF  = PREFIX_OFF + B_SZ;                   // 32
constexpr size_t KREM_OFF   = CLESS_OFF + B_SZ;                    // 32
constexpr size_t SUM_OFF    = KREM_OFF + B_SZ;                     // 32 (float)
constexpr size_t WS_WORDS   = SUM_OFF + B_SZ;

__global__ __launch_bounds__(THREADS)
void trim_init(unsigned* __restrict__ ws) {
    size_t i = (size_t)blockIdx.x * blockDim.x + threadIdx.x;
    if (i < WS_WORDS) {
        unsigned v = 0u;
        if (i >= KREM_OFF && i < KREM_OFF + B_SZ) v = K_SEL;  // remaining rank
        ws[i] = v;
    }
}

// One histogram level: count elements whose bits match cur_prefix under
// prefix_mask, binned by (bits >> shift) & (NBINS-1).
template <int NBINS>
__global__ __launch_bounds__(THREADS)
void trim_hist(const float* __restrict__ pred, const float* __restrict__ targ,
               const unsigned* __restrict__ cur_prefix,
               unsigned* __restrict__ ghist,
               unsigned prefix_mask, int shift) {
    __shared__ unsigned lhist[NBINS];
    const int b   = blockIdx.x / BLOCKS_PER_BATCH;
    const int blk = blockIdx.x % BLOCKS_PER_BATCH;
    for (int i = threadIdx.x; i < NBINS; i += THREADS) lhist[i] = 0u;
    __syncthreads();

    const unsigned pref = cur_prefix[b];                 // scalar (uniform)
    const size_t   base = (size_t)b * M_SZ + (size_t)blk * CHUNK;
    const float4*  p4 = (const float4*)(pred + base);
    const float4*  t4 = (const float4*)(targ + base);

    for (int i = threadIdx.x; i < CHUNK / 4; i += THREADS) {
        const float4 p = p4[i];
        const float4 t = t4[i];
        const float v[4] = { fabsf(p.x - t.x), fabsf(p.y - t.y),
                             fabsf(p.z - t.z), fabsf(p.w - t.w) };
#pragma unroll
        for (int j = 0; j < 4; ++j) {
            const unsigned bits = __float_as_uint(v[j]);
            if ((bits & prefix_mask) == pref) {
                atomicAdd(&lhist[(bits >> shift) & (NBINS - 1)], 1u);
            }
        }
    }
    __syncthreads();
    for (int i = threadIdx.x; i < NBINS; i += THREADS) {
        const unsigned c = lhist[i];
        if (c) atomicAdd(&ghist[(size_t)b * NBINS + i], c);
    }
}

// One wave32 per batch: each lane sums a contiguous segment of bins, a
// shfl_up exclusive prefix locates the segment containing the remaining
// rank, and only the owner lane walks its <=64 bins to find the digit.
__global__ __launch_bounds__(32)
void trim_scan(const unsigned* __restrict__ ghist, int nbins, int shift,
               unsigned* __restrict__ cur_prefix,
               unsigned* __restrict__ count_less,
               unsigned* __restrict__ krem) {
    const int b    = blockIdx.x;
    const int lane = threadIdx.x;                 // 0..31
    const unsigned need = krem[b];
    const int seg = nbins >> 5;                   // bins per lane (64 or 32)
    const unsigned* h = ghist + (size_t)b * nbins + (size_t)lane * seg;

    unsigned lsum = 0u;
    for (int i = 0; i < seg; ++i) lsum += h[i];

    // inclusive prefix across lanes -> exclusive
    unsigned incl = lsum;
#pragma unroll
    for (int off = 1; off < 32; off <<= 1) {
        const unsigned t = __shfl_up(incl, off, 32);
        if (lane >= off) incl += t;
    }
    const unsigned excl = incl - lsum;

    // unique owner: exclusive < need <= inclusive
    const bool owner = (excl < need) && (incl >= need);
    if (owner) {
        unsigned cum = excl;
        int d = lane * seg + seg - 1;
        for (int i = 0; i < seg; ++i) {
            const unsigned c = h[i];
            if (cum + c >= need) { d = lane * seg + i; break; }
            cum += c;
        }
        count_less[b] += cum;
        krem[b]        = need - cum;
        cur_prefix[b] |= ((unsigned)d) << shift;
    }
}

// Predicated sum of all values strictly below the threshold bit pattern.
// Wave32 reduction done with V_WMMA_F32_16X16X4_F32 (A = lane partials,
// B = ones): D[M][*] = p_M + p_{M+16}; sum 8 C VGPRs + one xor-16 shuffle.
__global__ __launch_bounds__(THREADS)
void trim_sum(const float* __restrict__ pred, const float* __restrict__ targ,
              const unsigned* __restrict__ thr_bits,
              float* __restrict__ sum_less) {
    const int b   = blockIdx.x / BLOCKS_PER_BATCH;
    const int blk = blockIdx.x % BLOCKS_PER_BATCH;
    const unsigned thr = thr_bits[b];
    const size_t  base = (size_t)b * M_SZ + (size_t)blk * CHUNK;
    const float4* p4 = (const float4*)(pred + base);
    const float4* t4 = (const float4*)(targ + base);

    float acc = 0.0f;
    for (int i = threadIdx.x; i < CHUNK / 4; i += THREADS) {
        const float4 p = p4[i];
        const float4 t = t4[i];
        const float v[4] = { fabsf(p.x - t.x), fabsf(p.y - t.y),
                             fabsf(p.z - t.z), fabsf(p.w - t.w) };
#pragma unroll
        for (int j = 0; j < 4; ++j) {
            const unsigned bits = __float_as_uint(v[j]);
            acc += (bits < thr) ? v[j] : 0.0f;   // select, no EXEC divergence
        }
    }

    // --- wave32 reduction via WMMA f32 16x16x4 (EXEC is all-ones here) ---
    v2f a;  a.x = acc;  a.y = 0.0f;      // A[M][K]: lane<16 K=0,1 ; lane>=16 K=2,3
    v2f bw; bw.x = 1.0f; bw.y = 1.0f;    // B = ones(4x16)
    v8f c = {0.0f, 0.0f, 0.0f, 0.0f, 0.0f, 0.0f, 0.0f, 0.0f};
    c = __builtin_amdgcn_wmma_f32_16x16x4_f32(
            /*neg_a=*/false, a, /*neg_b=*/false, bw,
            /*c_mod=*/(short)0, c, /*reuse_a=*/false, /*reuse_b=*/false);
    float s = c[0] + c[1] + c[2] + c[3] + c[4] + c[5] + c[6] + c[7];
    s += __shfl_xor(s, 16, 32);          // every lane now holds the wave total

    if ((threadIdx.x & 31) == 0) atomicAdd(&sum_less[b], s);
}

__global__ __launch_bounds__(32)
void trim_finalize(const float* __restrict__ sum_less,
                   const unsigned* __restrict__ count_less,
                   const unsigned* __restrict__ thr_bits,
                   float* __restrict__ out) {
    const int b = threadIdx.x;
    float contrib = 0.0f;
    if (b < B_SZ) {
        const float T = __uint_as_float(thr_bits[b]);
        contrib = sum_less[b] + (float)(int)(K_SEL - count_less[b]) * T;
    }
#pragma unroll
    for (int off = 16; off > 0; off >>= 1) contrib += __shfl_xor(contrib, off, 32);
    if (b == 0) out[0] = contrib / (2.0f * (float)M_SZ) / (float)B_SZ;
}

extern "C" void kernel_launch(void* const* d_in, const int* in_sizes, int n_in,
                              void* d_out, int out_size, void* d_ws, size_t ws_size,
                              hipStream_t stream) {
    (void)in_sizes; (void)n_in; (void)out_size; (void)ws_size;
    const float* pred = (const float*)d_in[0];
    const float* targ = (const float*)d_in[1];
    // d_in[2] (mask) is unused by the reference.
    float* out = (float*)d_out;

    unsigned* ws    = (unsigned*)d_ws;
    unsigned* hist0 = ws + HIST0_OFF;
    unsigned* hist1 = ws + HIST1_OFF;
    unsigned* hist2 = ws + HIST2_OFF;
    unsigned* pref  = ws + PREFIX_OFF;
    unsigned* cless = ws + CLESS_OFF;
    unsigned* krem  = ws + KREM_OFF;
    float*    sums  = (float*)(ws + SUM_OFF);

    const int grid = B_SZ * BLOCKS_PER_BATCH;   // 2400 blocks of 256
    const int init_blocks = (int)((WS_WORDS + THREADS - 1) / THREADS);

    trim_init<<<init_blocks, THREADS, 0, stream>>>(ws);

    // Level 0: bits [31:21]
    trim_hist<2048><<<grid, THREADS, 0, stream>>>(pred, targ, pref, hist0,
                                                  0x00000000u, 21);
    trim_scan<<<B_SZ, 32, 0, stream>>>(hist0, 2048, 21, pref, cless, krem);

    // Level 1: bits [20:10] within the chosen top digit
    trim_hist<2048><<<grid, THREADS, 0, stream>>>(pred, targ, pref, hist1,
                                                  0xFFE00000u, 10);
    trim_scan<<<B_SZ, 32, 0, stream>>>(hist1, 2048, 10, pref, cless, krem);

    // Level 2: bits [9:0] within the chosen 22-bit prefix
    trim_hist<1024><<<grid, THREADS, 0, stream>>>(pred, targ, pref, hist2,
                                                  0xFFFFFC00u, 0);
    trim_scan<<<B_SZ, 32, 0, stream>>>(hist2, 1024, 0, pref, cless, krem);

    // Sum all values strictly below the threshold, then combine.
    trim_sum<<<grid, THREADS, 0, stream>>>(pred, targ, pref, sums);
    trim_finalize<<<1, 32, 0, stream>>>(sums, cless, pref, out);
}